// GRAPH_AUTOENCODER_41918880809286
// MI455X (gfx1250) — compile-verified
//
#include <hip/hip_runtime.h>
#include <hip/hip_bf16.h>

typedef __attribute__((ext_vector_type(2))) float v2f;
typedef __attribute__((ext_vector_type(8))) float v8f;

#define BN_SCALE 0.999995000037f

// ---------------- problem constants ----------------
#define NN_  32768
#define EE_  524288
#define GG_  64
#define MM_  512
#define NP_  4096
#define MP_  64
#define EP_  32768

// ---------------- output offsets (floats) ----------------
#define OFF_Z       0L
#define OFF_ZGMLP   4194304L
#define OFF_ZNOISY  4202496L
#define OFF_XRECON  8396800L
#define OFF_A       12591104L
#define OFF_ZTILDE  29368320L
#define OFF_ZTGMLP  33562624L
#define OFF_POSG    33570816L
#define OFF_NEGG    33579008L
#define OFF_TGTZ    33587200L

// ---------------- workspace offsets (floats) ----------------
#define WS_NADJ   0L            // 64*512*512
#define WS_B1     16777216L     // N*256
#define WS_B2     25165824L
#define WS_B3     33554432L
#define WS_B4     41943040L
#define WS_DEG    50331648L     // N
#define WS_DINV   50364416L     // N
#define WS_DEGS   50397184L     // 4096
#define WS_DINVS  50401280L     // 4096
#define WS_ADJS   50405376L     // 64*64*64
#define WS_S1     50667520L     // 4096*256
#define WS_S2     51716096L
#define WS_S3     52764672L
#define WS_S4     53813248L
#define WS_ZG     54861824L     // 64*128
#define WS_T64A   54870016L     // 64*256
#define WS_T64B   54886400L
#define WS_DINV2  54902784L     // 64*512

// =====================================================================
// WMMA fp32 GEMM: one wave -> one 16x16 C tile, K stepped by 4.
//  TRANSB=false : C[m][n] = sum_k A[m][k]*B[k][n]   (adjacency @ features)
//  TRANSB=true  : C[m][n] = sum_k A[m][k]*B[n][k]   (h @ W^T, scores)
//  MODE 0: x(+bias)        1: relu(x+bias)   2: relu(x+bias)*BN_SCALE
//  MODE 3: relu(relu((x+bias)*BN_SCALE)+resid)      4: (x>0)?1:0
// =====================================================================
template<bool TRANSB, int MODE>
__global__ void k_gemm(const float* __restrict__ A, int lda, long sA,
                       const float* __restrict__ B, int ldb, long sB,
                       float* __restrict__ C, int ldc, long sC,
                       const float* __restrict__ bias,
                       const float* __restrict__ resid, long sR,
                       int M, int Ncols, int K)
{
    const int g = blockIdx.z;
    A += (long)g * sA;
    B += (long)g * sB;
    C += (long)g * sC;
    if (MODE == 3) resid += (long)g * sR;

    const int m0 = (blockIdx.x * blockDim.y + threadIdx.y) * 16;
    const int n0 = blockIdx.y * 16;
    if (m0 >= M) return;                 // wave-uniform: EXEC stays all-ones

    const int lane = threadIdx.x;        // 0..31 (wave32)
    const int lo = lane & 15;
    const int hi = lane >> 4;

    v8f acc = {0.f, 0.f, 0.f, 0.f, 0.f, 0.f, 0.f, 0.f};

    // A fragment (16x4 f32): lane%16 = M row, lane/16 selects K pair {0,1}/{2,3}
    const float* Ap = A + (long)(m0 + lo) * lda + 2 * hi;

    if (TRANSB) {
        const float* Bp = B + (long)(n0 + lo) * ldb + 2 * hi;
        for (int k0 = 0; k0 < K; k0 += 4) {
            v2f a, b;
            a.x = Ap[k0];  a.y = Ap[k0 + 1];
            b.x = Bp[k0];  b.y = Bp[k0 + 1];
            acc = __builtin_amdgcn_wmma_f32_16x16x4_f32(false, a, false, b,
                                                        (short)0, acc, false, false);
        }
    } else {
        for (int k0 = 0; k0 < K; k0 += 4) {
            v2f a, b;
            a.x = Ap[k0];  a.y = Ap[k0 + 1];
            const float* Bp = B + (long)(k0 + 2 * hi) * ldb + n0 + lo;
            b.x = Bp[0];   b.y = Bp[ldb];
            acc = __builtin_amdgcn_wmma_f32_16x16x4_f32(false, a, false, b,
                                                        (short)0, acc, false, false);
        }
    }

    const int col = n0 + lo;
    const float bv = (bias != nullptr) ? bias[col] : 0.0f;
#pragma unroll
    for (int v = 0; v < 8; ++v) {
        const int row = m0 + v + 8 * hi;   // C/D layout: VGPR v = rows v / v+8
        float x = acc[v] + bv;
        if (MODE == 1)      x = fmaxf(x, 0.0f);
        else if (MODE == 2) x = fmaxf(x, 0.0f) * BN_SCALE;
        else if (MODE == 3) {
            x = fmaxf(x * BN_SCALE, 0.0f);
            x = fmaxf(x + resid[(long)row * ldc + col], 0.0f);
        }
        else if (MODE == 4) x = (x > 0.0f) ? 1.0f : 0.0f;
        C[(long)row * ldc + col] = x;
    }
}

// ---------------- elementwise / scatter kernels ----------------
__global__ void k_fill(float* p, float v, long n) {
    long i = (long)blockIdx.x * blockDim.x + threadIdx.x;
    long st = (long)gridDim.x * blockDim.x;
    for (; i < n; i += st) p[i] = v;
}

__global__ void k_deg_scatter(const int* __restrict__ dst, float* deg, int E) {
    int i = blockIdx.x * blockDim.x + threadIdx.x;
    if (i < E) atomicAdd(&deg[dst[i]], 1.0f);
}

__global__ void k_rsqrt(const float* __restrict__ deg, float* dinv, int n) {
    int i = blockIdx.x * blockDim.x + threadIdx.x;
    if (i < n) dinv[i] = rsqrtf(fmaxf(deg[i], 1.0f));
}

// adj[(g*Mn+c)*Mn + r] layout (transposed adjacency: entry = edge r->c)
__global__ void k_adj_init(float* adj, const float* __restrict__ dinv, int Gn, int Mn) {
    long total = (long)Gn * Mn * Mn;
    long i = (long)blockIdx.x * blockDim.x + threadIdx.x;
    long st = (long)gridDim.x * blockDim.x;
    for (; i < total; i += st) {
        int r = (int)(i % Mn);
        int c = (int)((i / Mn) % Mn);
        int g = (int)(i / ((long)Mn * Mn));
        float d = dinv[g * Mn + c];
        adj[i] = (r == c) ? d * d : 0.0f;
    }
}

__global__ void k_adj_scatter(float* adj, const int* __restrict__ src,
                              const int* __restrict__ dst,
                              const float* __restrict__ dinv,
                              int E, int shift, int mask, int Mn) {
    int i = blockIdx.x * blockDim.x + threadIdx.x;
    if (i >= E) return;
    int s = src[i], d = dst[i];
    int g = s >> shift;
    int r = s & mask;
    int c = d & mask;
    atomicAdd(&adj[((long)(g * Mn + c)) * Mn + r], dinv[s] * dinv[d]);
}

// column degrees of A_hat (diag forced to 1), store rsqrt(max(deg,1))
__global__ void k_dinvA(const float* __restrict__ Aout, float* dinv2, int Gn, int Mn) {
    int idx = blockIdx.x * blockDim.x + threadIdx.x;
    if (idx >= Gn * Mn) return;
    int g = idx / Mn, c = idx % Mn;
    const float* Ag = Aout + (long)g * Mn * Mn;
    float s = 0.0f;
    for (int r = 0; r < Mn; ++r) s += Ag[(long)r * Mn + c];
    s = s - Ag[(long)c * Mn + c] + 1.0f;   // force diag to 1
    dinv2[idx] = rsqrtf(fmaxf(s, 1.0f));
}

__global__ void k_nadj2(const float* __restrict__ Aout, const float* __restrict__ dinv2,
                        float* adj, int Gn, int Mn) {
    long total = (long)Gn * Mn * Mn;
    long i = (long)blockIdx.x * blockDim.x + threadIdx.x;
    long st = (long)gridDim.x * blockDim.x;
    for (; i < total; i += st) {
        int r = (int)(i % Mn);
        int c = (int)((i / Mn) % Mn);
        int g = (int)(i / ((long)Mn * Mn));
        float a = (r == c) ? 1.0f : Aout[((long)g * Mn + r) * Mn + c];
        adj[i] = a * dinv2[g * Mn + r] * dinv2[g * Mn + c];
    }
}

// l2 normalize 128-wide rows in place; optionally mirror to out1/out2
__global__ void k_l2norm128(float* h, float* o1, float* o2) {
    __shared__ float sh[128];
    long row = blockIdx.x;
    int t = threadIdx.x;
    float v = h[row * 128 + t];
    sh[t] = v * v;
    __syncthreads();
    for (int s = 64; s > 0; s >>= 1) {
        if (t < s) sh[t] += sh[t + s];
        __syncthreads();
    }
    float n = sqrtf(sh[0]);
    float r = v / fmaxf(n, 1e-12f);
    h[row * 128 + t] = r;
    if (o1) o1[row * 128 + t] = r;
    if (o2) o2[row * 128 + t] = r;
}

__global__ void k_maxpool(const float* __restrict__ h, float* out, int Gn, int Mn, int D) {
    int idx = blockIdx.x * blockDim.x + threadIdx.x;
    if (idx >= Gn * D) return;
    int g = idx / D, d = idx % D;
    const float* p = h + (long)g * Mn * D + d;
    float m = p[0];
    for (int r = 1; r < Mn; ++r) m = fmaxf(m, p[(long)r * D]);
    out[idx] = m;
}

__global__ void k_meanpool(const float* __restrict__ h, float* out, int Gn, int Mn, int D) {
    int idx = blockIdx.x * blockDim.x + threadIdx.x;
    if (idx >= Gn * D) return;
    int g = idx / D, d = idx % D;
    const float* p = h + (long)g * Mn * D + d;
    float s = 0.0f;
    for (int r = 0; r < Mn; ++r) s += p[(long)r * D];
    out[idx] = s / (float)Mn;
}

// train-mode BatchNorm1d over rows (batch stats), in place
__global__ void k_bn_cols(float* h, int rows, int cols) {
    int c = blockIdx.x * blockDim.x + threadIdx.x;
    if (c >= cols) return;
    float mu = 0.0f;
    for (int r = 0; r < rows; ++r) mu += h[(long)r * cols + c];
    mu /= (float)rows;
    float var = 0.0f;
    for (int r = 0; r < rows; ++r) {
        float d = h[(long)r * cols + c] - mu;
        var += d * d;
    }
    var /= (float)rows;
    float inv = rsqrtf(var + 1e-5f);
    for (int r = 0; r < rows; ++r)
        h[(long)r * cols + c] = (h[(long)r * cols + c] - mu) * inv;
}

// ---------------- host-side GEMM dispatcher ----------------
static void gemm(hipStream_t st, bool transB, int mode,
                 const float* A, int lda, long sA,
                 const float* B, int ldb, long sB,
                 float* C, int ldc, long sC,
                 const float* bias, const float* resid, long sR,
                 int M, int Nc, int K, int batch)
{
    dim3 block(32, 4, 1);
    dim3 grid((M / 16 + 3) / 4, Nc / 16, batch);
    if (transB) {
        switch (mode) {
        case 0: k_gemm<true, 0><<<grid, block, 0, st>>>(A, lda, sA, B, ldb, sB, C, ldc, sC, bias, resid, sR, M, Nc, K); break;
        case 1: k_gemm<true, 1><<<grid, block, 0, st>>>(A, lda, sA, B, ldb, sB, C, ldc, sC, bias, resid, sR, M, Nc, K); break;
        case 2: k_gemm<true, 2><<<grid, block, 0, st>>>(A, lda, sA, B, ldb, sB, C, ldc, sC, bias, resid, sR, M, Nc, K); break;
        default: k_gemm<true, 4><<<grid, block, 0, st>>>(A, lda, sA, B, ldb, sB, C, ldc, sC, bias, resid, sR, M, Nc, K); break;
        }
    } else {
        switch (mode) {
        case 0: k_gemm<false, 0><<<grid, block, 0, st>>>(A, lda, sA, B, ldb, sB, C, ldc, sC, bias, resid, sR, M, Nc, K); break;
        default: k_gemm<false, 3><<<grid, block, 0, st>>>(A, lda, sA, B, ldb, sB, C, ldc, sC, bias, resid, sR, M, Nc, K); break;
        }
    }
}

// ---------------- one sparse-GCN subgraph encoder (pos / neg) ----------------
static void run_subgraph(hipStream_t st, float* ws,
                         const float* sx, const int* ssrc, const int* sdst,
                         const float* W0, const float* b0, const float* sW0, const float* sb0,
                         const float* W1, const float* b1,
                         const float* W2, const float* b2, const float* sW2, const float* sb2,
                         float* outg)
{
    float* degS = ws + WS_DEGS;
    float* dinvS = ws + WS_DINVS;
    float* adjS = ws + WS_ADJS;
    float* s1 = ws + WS_S1;
    float* s2 = ws + WS_S2;
    float* s3 = ws + WS_S3;
    float* s4 = ws + WS_S4;

    k_fill<<<16, 256, 0, st>>>(degS, 1.0f, NP_);
    k_deg_scatter<<<(EP_ + 255) / 256, 256, 0, st>>>(sdst, degS, EP_);
    k_rsqrt<<<(NP_ + 255) / 256, 256, 0, st>>>(degS, dinvS, NP_);
    k_adj_init<<<512, 256, 0, st>>>(adjS, dinvS, GG_, MP_);
    k_adj_scatter<<<(EP_ + 255) / 256, 256, 0, st>>>(adjS, ssrc, sdst, dinvS, EP_, 6, 63, MP_);

    // layer 1: 128 -> 256 (+shortcut)
    gemm(st, true, 0, sx, 128, 0, W0, 128, 0, s1, 256, 0, nullptr, nullptr, 0, NP_, 256, 128, 1);
    gemm(st, true, 0, sx, 128, 0, sW0, 128, 0, s2, 256, 0, sb0, nullptr, 0, NP_, 256, 128, 1);
    gemm(st, false, 3, adjS, MP_, (long)MP_ * MP_, s1, 256, (long)MP_ * 256,
         s3, 256, (long)MP_ * 256, b0, s2, (long)MP_ * 256, MP_, 256, MP_, GG_);
    // layer 2: 256 -> 256 (identity resid)
    gemm(st, true, 0, s3, 256, 0, W1, 256, 0, s1, 256, 0, nullptr, nullptr, 0, NP_, 256, 256, 1);
    gemm(st, false, 3, adjS, MP_, (long)MP_ * MP_, s1, 256, (long)MP_ * 256,
         s4, 256, (long)MP_ * 256, b1, s3, (long)MP_ * 256, MP_, 256, MP_, GG_);
    // layer 3: 256 -> 128 (+shortcut)
    gemm(st, true, 0, s4, 256, 0, sW2, 256, 0, s2, 128, 0, sb2, nullptr, 0, NP_, 128, 256, 1);
    gemm(st, true, 0, s4, 256, 0, W2, 256, 0, s1, 128, 0, nullptr, nullptr, 0, NP_, 128, 256, 1);
    gemm(st, false, 3, adjS, MP_, (long)MP_ * MP_, s1, 128, (long)MP_ * 128,
         s3, 128, (long)MP_ * 128, b2, s2, (long)MP_ * 128, MP_, 128, MP_, GG_);

    k_l2norm128<<<NP_, 128, 0, st>>>(s3, nullptr, nullptr);
    k_meanpool<<<(GG_ * 128 + 255) / 256, 256, 0, st>>>(s3, outg, GG_, MP_, 128);
}

// =====================================================================
extern "C" void kernel_launch(void* const* d_in, const int* in_sizes, int n_in,
                              void* d_out, int out_size, void* d_ws, size_t ws_size,
                              hipStream_t stream)
{
    const float* x = (const float*)d_in[0];
    const int* ei = (const int*)d_in[1];
    const int* src = ei;
    const int* dst = ei + EE_;
    const float* posx = (const float*)d_in[4];
    const int* pei = (const int*)d_in[5];
    const float* negx = (const float*)d_in[7];
    const int* nei = (const int*)d_in[8];
    const float* tx = (const float*)d_in[10];

    // params, pytree (sorted-key) order
    const float* decW0 = (const float*)d_in[11]; const float* decb0 = (const float*)d_in[12];
    const float* decW1 = (const float*)d_in[13]; const float* decb1 = (const float*)d_in[14];
    const float* decW2 = (const float*)d_in[15]; const float* decb2 = (const float*)d_in[16];
    const float* decW3 = (const float*)d_in[17]; const float* decb3 = (const float*)d_in[18];
    const float* edgeW = (const float*)d_in[19];
    const float* eW0 = (const float*)d_in[20]; const float* eb0 = (const float*)d_in[21];
    const float* esW0 = (const float*)d_in[22]; const float* esb0 = (const float*)d_in[23];
    const float* eW1 = (const float*)d_in[24]; const float* eb1 = (const float*)d_in[25];
    const float* eW2 = (const float*)d_in[26]; const float* eb2 = (const float*)d_in[27];
    const float* esW2 = (const float*)d_in[28]; const float* esb2 = (const float*)d_in[29];
    const float* uW0 = (const float*)d_in[30]; const float* ub0 = (const float*)d_in[31];
    const float* usW0 = (const float*)d_in[32]; const float* usb0 = (const float*)d_in[33];
    const float* uW1 = (const float*)d_in[34]; const float* ub1 = (const float*)d_in[35];
    const float* uW2 = (const float*)d_in[36]; const float* ub2 = (const float*)d_in[37];
    const float* usW2 = (const float*)d_in[38]; const float* usb2 = (const float*)d_in[39];
    const float* nW0 = (const float*)d_in[40]; const float* nb0 = (const float*)d_in[41];
    const float* nW1 = (const float*)d_in[42]; const float* nb1 = (const float*)d_in[43];
    const float* nW2 = (const float*)d_in[44]; const float* nb2 = (const float*)d_in[45];
    const float* pW1 = (const float*)d_in[46]; const float* pb1 = (const float*)d_in[47];
    const float* pW2 = (const float*)d_in[48]; const float* pb2 = (const float*)d_in[49];

    float* out = (float*)d_out;
    float* ws = (float*)d_ws;
    float* adj = ws + WS_NADJ;
    float* b1 = ws + WS_B1;
    float* b2 = ws + WS_B2;
    float* b3 = ws + WS_B3;
    float* b4 = ws + WS_B4;
    float* deg = ws + WS_DEG;
    float* dinv = ws + WS_DINV;
    float* zg = ws + WS_ZG;
    float* t64a = ws + WS_T64A;
    float* t64b = ws + WS_T64B;
    float* dinv2 = ws + WS_DINV2;

    const long gMM = (long)MM_ * MM_;    // per-graph adjacency stride
    const long g256 = (long)MM_ * 256;
    const long g128 = (long)MM_ * 128;

    // ---- degrees + normalized (transposed) adjacency of the main graph ----
    k_fill<<<256, 256, 0, stream>>>(deg, 1.0f, NN_);
    k_deg_scatter<<<(EE_ + 255) / 256, 256, 0, stream>>>(dst, deg, EE_);
    k_rsqrt<<<(NN_ + 255) / 256, 256, 0, stream>>>(deg, dinv, NN_);
    k_adj_init<<<8192, 256, 0, stream>>>(adj, dinv, GG_, MM_);
    k_adj_scatter<<<(EE_ + 255) / 256, 256, 0, stream>>>(adj, src, dst, dinv, EE_, 9, 511, MM_);

    // ---- sparse encoder (as dense batched GEMM) ----
    // layer 1: 128 -> 256, shortcut
    gemm(stream, true, 0, x, 128, 0, eW0, 128, 0, b1, 256, 0, nullptr, nullptr, 0, NN_, 256, 128, 1);
    gemm(stream, true, 0, x, 128, 0, esW0, 128, 0, b2, 256, 0, esb0, nullptr, 0, NN_, 256, 128, 1);
    gemm(stream, false, 3, adj, MM_, gMM, b1, 256, g256, b3, 256, g256, eb0, b2, g256, MM_, 256, MM_, GG_);
    // layer 2: 256 -> 256
    gemm(stream, true, 0, b3, 256, 0, eW1, 256, 0, b1, 256, 0, nullptr, nullptr, 0, NN_, 256, 256, 1);
    gemm(stream, false, 3, adj, MM_, gMM, b1, 256, g256, b4, 256, g256, eb1, b3, g256, MM_, 256, MM_, GG_);
    // layer 3: 256 -> 128, shortcut
    gemm(stream, true, 0, b4, 256, 0, esW2, 256, 0, b2, 128, 0, esb2, nullptr, 0, NN_, 128, 256, 1);
    gemm(stream, true, 0, b4, 256, 0, eW2, 256, 0, b1, 128, 0, nullptr, nullptr, 0, NN_, 128, 256, 1);
    gemm(stream, false, 3, adj, MM_, gMM, b1, 128, g128, b3, 128, g128, eb2, b2, g128, MM_, 128, MM_, GG_);
    // z = l2norm(h3)  (also fills z_noisy, identical since noise_std == 0)
    k_l2norm128<<<NN_, 128, 0, stream>>>(b3, out + OFF_Z, out + OFF_ZNOISY);

    // ---- bilinear edge decoder: A = (z @ edge_W @ z^T > 0) ----
    gemm(stream, false, 0, b3, 128, 0, edgeW, 128, 0, b1, 128, 0, nullptr, nullptr, 0, NN_, 128, 128, 1);
    gemm(stream, true, 4, b1, 128, g128, b3, 128, g128, out + OFF_A, MM_, gMM,
         nullptr, nullptr, 0, MM_, MM_, 128, GG_);
    // rebuild normalized adjacency from A_hat (diag forced to 1), reuse adj buffer
    k_dinvA<<<(GG_ * MM_ + 255) / 256, 256, 0, stream>>>(out + OFF_A, dinv2, GG_, MM_);
    k_nadj2<<<8192, 256, 0, stream>>>(out + OFF_A, dinv2, adj, GG_, MM_);

    // ---- feature decoder: 128 -> 128 -> 256 -> 256 -> 128 ----
    gemm(stream, true, 2, b3, 128, 0, decW0, 128, 0, b1, 128, 0, decb0, nullptr, 0, NN_, 128, 128, 1);
    gemm(stream, true, 2, b1, 128, 0, decW1, 128, 0, b2, 256, 0, decb1, nullptr, 0, NN_, 256, 128, 1);
    gemm(stream, true, 2, b2, 256, 0, decW2, 256, 0, b1, 256, 0, decb2, nullptr, 0, NN_, 256, 256, 1);
    gemm(stream, true, 0, b1, 256, 0, decW3, 256, 0, b4, 128, 0, decb3, nullptr, 0, NN_, 128, 256, 1);
    hipMemcpyAsync(out + OFF_XRECON, b4, (size_t)NN_ * 128 * sizeof(float),
                   hipMemcpyDeviceToDevice, stream);

    // ---- z_g = per-graph max pool, then projection head ----
    k_maxpool<<<(GG_ * 128 + 255) / 256, 256, 0, stream>>>(b3, zg, GG_, MM_, 128);
    gemm(stream, true, 1, zg, 128, 0, pW1, 128, 0, t64a, 128, 0, pb1, nullptr, 0, GG_, 128, 128, 1);
    gemm(stream, true, 0, t64a, 128, 0, pW2, 128, 0, out + OFF_ZGMLP, 128, 0, pb2, nullptr, 0, GG_, 128, 128, 1);

    // ---- re-encode x_recon over rebuilt adjacency (same enc weights) ----
    gemm(stream, true, 0, b4, 128, 0, eW0, 128, 0, b1, 256, 0, nullptr, nullptr, 0, NN_, 256, 128, 1);
    gemm(stream, true, 0, b4, 128, 0, esW0, 128, 0, b2, 256, 0, esb0, nullptr, 0, NN_, 256, 128, 1);
    gemm(stream, false, 3, adj, MM_, gMM, b1, 256, g256, b3, 256, g256, eb0, b2, g256, MM_, 256, MM_, GG_);
    gemm(stream, true, 0, b3, 256, 0, eW1, 256, 0, b1, 256, 0, nullptr, nullptr, 0, NN_, 256, 256, 1);
    gemm(stream, false, 3, adj, MM_, gMM, b1, 256, g256, b2, 256, g256, eb1, b3, g256, MM_, 256, MM_, GG_);
    gemm(stream, true, 0, b2, 256, 0, esW2, 256, 0, b4, 128, 0, esb2, nullptr, 0, NN_, 128, 256, 1);
    gemm(stream, true, 0, b2, 256, 0, eW2, 256, 0, b1, 128, 0, nullptr, nullptr, 0, NN_, 128, 256, 1);
    gemm(stream, false, 3, adj, MM_, gMM, b1, 128, g128, b3, 128, g128, eb2, b4, g128, MM_, 128, MM_, GG_);
    k_l2norm128<<<NN_, 128, 0, stream>>>(b3, out + OFF_ZTILDE, nullptr);
    k_maxpool<<<(GG_ * 128 + 255) / 256, 256, 0, stream>>>(b3, zg, GG_, MM_, 128);
    gemm(stream, true, 1, zg, 128, 0, pW1, 128, 0, t64a, 128, 0, pb1, nullptr, 0, GG_, 128, 128, 1);
    gemm(stream, true, 0, t64a, 128, 0, pW2, 128, 0, out + OFF_ZTGMLP, 128, 0, pb2, nullptr, 0, GG_, 128, 128, 1);

    // ---- pos / neg subgraph encoders (enc_sub weights) ----
    run_subgraph(stream, ws, posx, pei, pei + EP_,
                 uW0, ub0, usW0, usb0, uW1, ub1, uW2, ub2, usW2, usb2, out + OFF_POSG);
    run_subgraph(stream, ws, negx, nei, nei + EP_,
                 uW0, ub0, usW0, usb0, uW1, ub1, uW2, ub2, usW2, usb2, out + OFF_NEGG);

    // ---- target node encoder: Linear+ReLU+BN, Linear+ReLU+BN, Linear+ReLU, l2norm ----
    gemm(stream, true, 1, tx, 128, 0, nW0, 128, 0, t64a, 256, 0, nb0, nullptr, 0, GG_, 256, 128, 1);
    k_bn_cols<<<1, 256, 0, stream>>>(t64a, GG_, 256);
    gemm(stream, true, 1, t64a, 256, 0, nW1, 256, 0, t64b, 256, 0, nb1, nullptr, 0, GG_, 256, 256, 1);
    k_bn_cols<<<1, 256, 0, stream>>>(t64b, GG_, 256);
    gemm(stream, true, 1, t64b, 256, 0, nW2, 256, 0, t64a, 128, 0, nb2, nullptr, 0, GG_, 128, 256, 1);
    k_l2norm128<<<GG_, 128, 0, stream>>>(t64a, out + OFF_TGTZ, nullptr);
}